// MyModel_87522843559467
// MI455X (gfx1250) — compile-verified
//
#include <hip/hip_runtime.h>
#include <math.h>

#define VOCAB 100000
#define SENT  256
#define DIM   128
#define H1    20
#define H2    5
#define W1PAD 32   // zero-padded N extent for the 128x20 weight tile in LDS

typedef __attribute__((ext_vector_type(2))) float v2f;
typedef __attribute__((ext_vector_type(4))) float v4f;
typedef __attribute__((ext_vector_type(8))) float v8f;

// ---------------------------------------------------------------------------
// Kernel 1: recover word indices from the one-hot matrix.
// Streams 102.4 MB fully coalesced (b128 loads, non-temporal); the single
// nonzero per row produces one scattered int store (256 total).
// ---------------------------------------------------------------------------
__global__ void find_indices(const float* __restrict__ oh, int* __restrict__ wid) {
  const long total4 = (long)SENT * VOCAB / 4;   // VOCAB % 4 == 0 -> vec4 never straddles a row
  const v4f* oh4 = (const v4f*)oh;
  long i = (long)blockIdx.x * blockDim.x + threadIdx.x;
  const long stride = (long)gridDim.x * blockDim.x;
  for (; i < total4; i += stride) {
    v4f v = __builtin_nontemporal_load(oh4 + i);
    if (v[0] != 0.0f || v[1] != 0.0f || v[2] != 0.0f || v[3] != 0.0f) {
      long idx0 = i * 4;
      int s  = (int)(idx0 / VOCAB);
      int v0 = (int)(idx0 - (long)s * VOCAB);
      if (v[0] != 0.0f) wid[s] = v0 + 0;
      if (v[1] != 0.0f) wid[s] = v0 + 1;
      if (v[2] != 0.0f) wid[s] = v0 + 2;
      if (v[3] != 0.0f) wid[s] = v0 + 3;
    }
  }
}

// ---------------------------------------------------------------------------
// Kernel 2: gather-sum 256 embedding rows (128 KB instead of a 51.2 MB dense
// GEMV), then dense(128->20)+ReLU via V_WMMA_F32_16X16X4_F32 on wave 0 with a
// branch-free LDS-resident inner loop, then dense(20->5)+softmax on lane 0.
// ---------------------------------------------------------------------------
__global__ void head_kernel(const float* __restrict__ WV,
                            const float* __restrict__ W1,
                            const float* __restrict__ b1,
                            const float* __restrict__ W2,
                            const float* __restrict__ b2,
                            const int*   __restrict__ wid,
                            float* __restrict__ out) {
  __shared__ int   swid[SENT];
  __shared__ float fs2[2 * DIM];
  __shared__ float fs[DIM];
  __shared__ float sW1[DIM * W1PAD];    // 16 KB, columns 20..31 zeroed
  __shared__ float xs[H1];

  const int t = threadIdx.x;            // 256 threads = 8 waves (wave32)
  swid[t] = wid[t];

  // Stage W1 into LDS, zero-padded to 128x32 so the WMMA loop is branch-free.
  for (int i = t; i < DIM * W1PAD; i += SENT) {
    const int r = i >> 5;               // 0..127 (K)
    const int c = i & (W1PAD - 1);      // 0..31  (N)
    sW1[i] = (c < H1) ? W1[r * H1 + c] : 0.0f;
  }
  __syncthreads();

  // Phase 1: feature_sum[d] = sum_s WV[wid[s]][d]; 2 partial halves, coalesced.
  const int d    = t & (DIM - 1);
  const int part = t >> 7;
  float sum = 0.0f;
  const int s0 = part * 128;
  for (int s = s0; s < s0 + 128; ++s)
    sum += WV[(long)swid[s] * DIM + d];
  fs2[t] = sum;
  __syncthreads();
  if (t < DIM) fs[t] = fs2[t] + fs2[t + DIM];
  __syncthreads();

  // Phase 2: x = relu(fs @ W1 + b1) with fp32 WMMA on wave 0 (EXEC all-1s,
  // no branches inside the loop -> pure ds_load + v_wmma).
  if ((t >> 5) == 0) {
    const int lane  = t;                // 0..31
    const int col   = lane & 15;        // N within a 16-wide tile
    const int khalf = lane >> 4;        // 0: K={0,1}, 1: K={2,3} (ISA A/B layout)

    v8f c0 = {};                        // N tile 0: cols 0..15
    v8f c1 = {};                        // N tile 1: cols 16..19 (zero-padded)

    for (int k0 = 0; k0 < DIM; k0 += 4) {
      const int kA = k0 + 2 * khalf;
      // A 16x4, broadcast the same row into all 16 M rows:
      v2f a;
      a[0] = fs[kA + 0];
      a[1] = fs[kA + 1];
      // B 4x16 tiles from LDS-staged, padded W1:
      v2f b0, b1v;
      b0[0]  = sW1[(kA + 0) * W1PAD + col];
      b0[1]  = sW1[(kA + 1) * W1PAD + col];
      b1v[0] = sW1[(kA + 0) * W1PAD + 16 + col];
      b1v[1] = sW1[(kA + 1) * W1PAD + 16 + col];

      c0 = __builtin_amdgcn_wmma_f32_16x16x4_f32(false, a, false, b0,
                                                 (short)0, c0, false, false);
      c1 = __builtin_amdgcn_wmma_f32_16x16x4_f32(false, a, false, b1v,
                                                 (short)0, c1, false, false);
    }

    // D row 0 lives in element 0 of lanes 0..15 (M=0, N=lane).
    if (lane < 16) {
      float v = c0[0] + b1[lane];
      xs[lane] = v > 0.0f ? v : 0.0f;
    }
    if (lane < H1 - 16) {
      float v = c1[0] + b1[16 + lane];
      xs[16 + lane] = v > 0.0f ? v : 0.0f;
    }

    // Phase 3: logits = xs @ W2 + b2; softmax. Tiny -> serial on lane 0.
    if (lane == 0) {
      float logits[H2];
      float mx = -1e30f;
      for (int i = 0; i < H2; ++i) {
        float acc = b2[i];
        for (int j = 0; j < H1; ++j) acc += xs[j] * W2[j * H2 + i];
        logits[i] = acc;
        if (acc > mx) mx = acc;
      }
      float den = 0.0f;
      for (int i = 0; i < H2; ++i) { logits[i] = __expf(logits[i] - mx); den += logits[i]; }
      for (int i = 0; i < H2; ++i) out[i] = logits[i] / den;
    }
  }
}

// ---------------------------------------------------------------------------
extern "C" void kernel_launch(void* const* d_in, const int* in_sizes, int n_in,
                              void* d_out, int out_size, void* d_ws, size_t ws_size,
                              hipStream_t stream) {
  const float* oh = (const float*)d_in[0];   // (256, 100000) f32
  const float* WV = (const float*)d_in[1];   // (100000, 128) f32
  const float* W1 = (const float*)d_in[2];   // (128, 20) f32
  const float* b1 = (const float*)d_in[3];   // (20,) f32
  const float* W2 = (const float*)d_in[4];   // (20, 5) f32
  const float* b2 = (const float*)d_in[5];   // (5,) f32
  (void)in_sizes; (void)n_in; (void)out_size; (void)ws_size;

  int* wid = (int*)d_ws;                     // 256 ints of scratch

  // Saturate HBM on the 102.4 MB one-hot stream.
  find_indices<<<4096, 256, 0, stream>>>(oh, wid);
  // Gather + MLP head (tiny; single workgroup).
  head_kernel<<<1, 256, 0, stream>>>(WV, W1, b1, W2, b2, wid, (float*)d_out);
}